// Hybrid_Loss_86887188398258
// MI455X (gfx1250) — compile-verified
//
#include <hip/hip_runtime.h>
#include <hip/hip_bf16.h>
#include <math.h>

typedef __attribute__((ext_vector_type(2))) float v2f;
typedef __attribute__((ext_vector_type(8))) float v8f;

#define DIM 128
#define MARGIN 0.5f
#define L_CONST 200.0f

// ---------------------------------------------------------------------------
// Fused triplet-hard kernel: one wave per 16-row tile of the distance matrix.
// blockIdx.y selects the target matrix (0 = opt_feat, 1 = semi_feat).
// Streams over all column tiles, doing  S = A * B^T  with V_WMMA_F32_16X16X4_F32
// and keeping only running hardest-positive / hardest-negative per row.
// ---------------------------------------------------------------------------
__global__ __launch_bounds__(32) void triplet_hard_kernel(
    const float* __restrict__ src, const float* __restrict__ tgt0,
    const float* __restrict__ tgt1, const int* __restrict__ labels,
    float* __restrict__ acc, int n) {
  const float* tgt = (blockIdx.y == 0) ? tgt0 : tgt1;
  const int i0   = blockIdx.x * 16;
  const int lane = threadIdx.x;      // 0..31
  const int half = lane >> 4;        // 0 or 1 (selects K sub-pair per WMMA layout)
  const int l16  = lane & 15;

  // --- Load A fragments (16 rows x 128 K) in WMMA 16x16x4 f32 A layout ---
  // lane l16 (half h): row = i0+l16, per kb holds K = {4kb+2h, 4kb+2h+1}
  v2f afrag[DIM / 4];
  const float* aptr = src + (size_t)(i0 + l16) * DIM + half * 2;
  float a2p = 0.0f;
#pragma unroll
  for (int kb = 0; kb < DIM / 4; ++kb) {
    afrag[kb] = *(const v2f*)(aptr + kb * 4);
    a2p += afrag[kb].x * afrag[kb].x + afrag[kb].y * afrag[kb].y;
  }
  // combine the two K-halves: lane l16 (both halves) now holds ||src[i0+l16]||^2
  float a2 = a2p + __shfl_xor(a2p, 16);
  int mylab = labels[i0 + l16];

  float ap[8], an[8];
#pragma unroll
  for (int r = 0; r < 8; ++r) { ap[r] = -INFINITY; an[r] = INFINITY; }

  for (int c0 = 0; c0 < n; c0 += 16) {
    // --- Batch-load the B column tile (16 cols x 128 K) into registers ---
    // Issued as one load clause so waits amortize instead of per-fragment stalls.
    const float* bptr = tgt + (size_t)(c0 + l16) * DIM + half * 2;
    v2f bfrag[DIM / 4];
#pragma unroll
    for (int kb = 0; kb < DIM / 4; ++kb) {
      bfrag[kb] = *(const v2f*)(bptr + kb * 4);
    }
    // Prefetch next column tile while this tile's WMMAs run.
    if (c0 + 16 < n) {
      __builtin_prefetch(tgt + (size_t)(c0 + 16 + l16) * DIM, 0, 3);
    }

    // --- WMMA chain with two accumulators to break the RAW dependency chain ---
    v8f acc0 = {};
    v8f acc1 = {};
    float b2p = 0.0f;
#pragma unroll
    for (int kb = 0; kb < DIM / 4; kb += 2) {
      acc0 = __builtin_amdgcn_wmma_f32_16x16x4_f32(
          false, afrag[kb], false, bfrag[kb], (short)0, acc0, false, false);
      acc1 = __builtin_amdgcn_wmma_f32_16x16x4_f32(
          false, afrag[kb + 1], false, bfrag[kb + 1], (short)0, acc1, false, false);
      b2p += bfrag[kb].x * bfrag[kb].x + bfrag[kb].y * bfrag[kb].y;
      b2p += bfrag[kb + 1].x * bfrag[kb + 1].x + bfrag[kb + 1].y * bfrag[kb + 1].y;
    }
    float b2 = b2p + __shfl_xor(b2p, 16);   // ||tgt[c0+l16]||^2
    int collab = labels[c0 + l16];

#pragma unroll
    for (int r = 0; r < 8; ++r) {
      const int M = half * 8 + r;           // row index within tile (D layout)
      float a2m  = __shfl(a2, M);           // ||row||^2 lives in lane M (<16)
      int   labm = __shfl(mylab, M);
      float s_mn = acc0[r] + acc1[r];
      float d = fmaxf(a2m + b2 - 2.0f * s_mn, 0.0f);
      bool same = (labm == collab);
      ap[r] = same ? fmaxf(ap[r], d) : ap[r];
      an[r] = same ? an[r] : fminf(an[r], d);
    }
  }

  // --- Reduce over N: lanes within each 16-lane half share the same M set ---
#pragma unroll
  for (int m = 1; m < 16; m <<= 1) {
#pragma unroll
    for (int r = 0; r < 8; ++r) {
      ap[r] = fmaxf(ap[r], __shfl_xor(ap[r], m));
      an[r] = fminf(an[r], __shfl_xor(an[r], m));
    }
  }

  float s = 0.0f;
#pragma unroll
  for (int r = 0; r < 8; ++r) s += fmaxf(ap[r] - an[r] + MARGIN, 0.0f);
  s += __shfl_xor(s, 16);                   // rows 0..7 (half 0) + 8..15 (half 1)
  if (lane == 0) atomicAdd(acc, s);
}

// ---------------------------------------------------------------------------
// IoU-ratio loss: one wave per row; float4 loads, shfl reductions.
// ---------------------------------------------------------------------------
__device__ __forceinline__ float iou_score(float dx, float dy) {
  float ax = fabsf(dx) * (1.0f / L_CONST);
  float ay = fabsf(dy) * (1.0f / L_CONST);
  return 1.0f / (1.0f - (1.0f - ax) * (1.0f - ay) * 0.5f) - 1.0f;
}

__global__ __launch_bounds__(256) void iou_loss_kernel(
    const float* __restrict__ sar, const float* __restrict__ opt,
    const float* __restrict__ semi, const float* __restrict__ sar_pos,
    const float* __restrict__ opt_pos, const float* __restrict__ semi_pos,
    float* __restrict__ acc, int n) {
  const int row  = (blockIdx.x * blockDim.x + threadIdx.x) >> 5;
  const int lane = threadIdx.x & 31;
  if (row >= n) return;

  const float4 a = ((const float4*)(sar  + (size_t)row * DIM))[lane];
  const float4 b = ((const float4*)(opt  + (size_t)row * DIM))[lane];
  const float4 c = ((const float4*)(semi + (size_t)row * DIM))[lane];
  float s1 = a.x * b.x + a.y * b.y + a.z * b.z + a.w * b.w;
  float s2 = a.x * c.x + a.y * c.y + a.z * c.z + a.w * c.w;
#pragma unroll
  for (int m = 1; m < 32; m <<= 1) {
    s1 += __shfl_xor(s1, m);
    s2 += __shfl_xor(s2, m);
  }
  if (lane == 0) {
    float dx1 = opt_pos[2 * row]     - sar_pos[2 * row];
    float dy1 = opt_pos[2 * row + 1] - sar_pos[2 * row + 1];
    float dx2 = semi_pos[2 * row]     - sar_pos[2 * row];
    float dy2 = semi_pos[2 * row + 1] - sar_pos[2 * row + 1];
    float ratio = iou_score(dx2, dy2) / iou_score(dx1, dy1);
    float err = s2 / s1 - ratio;
    atomicAdd(acc, err * err);
  }
}

// ---------------------------------------------------------------------------
__global__ void init_kernel(float* ws) {
  ws[0] = 0.0f;   // sum of relu terms (both triplet losses)
  ws[1] = 0.0f;   // sum of squared iou errors
}

__global__ void finalize_kernel(const float* ws, float* out, float inv_n) {
  out[0] = (ws[0] + ws[1]) * inv_n;
}

// ---------------------------------------------------------------------------
extern "C" void kernel_launch(void* const* d_in, const int* in_sizes, int n_in,
                              void* d_out, int out_size, void* d_ws, size_t ws_size,
                              hipStream_t stream) {
  const float* sar      = (const float*)d_in[0];
  const float* opt      = (const float*)d_in[1];
  const float* semi     = (const float*)d_in[2];
  const float* sar_pos  = (const float*)d_in[3];
  const float* opt_pos  = (const float*)d_in[4];
  const float* semi_pos = (const float*)d_in[5];
  const int*   labels   = (const int*)d_in[6];

  const int n = in_sizes[6];          // 4096
  float* ws  = (float*)d_ws;
  float* out = (float*)d_out;

  init_kernel<<<1, 1, 0, stream>>>(ws);

  dim3 tg(n / 16, 2);                 // 256 row-tiles x {opt, semi}
  triplet_hard_kernel<<<tg, 32, 0, stream>>>(sar, opt, semi, labels, ws, n);

  iou_loss_kernel<<<(n * 32 + 255) / 256, 256, 0, stream>>>(
      sar, opt, semi, sar_pos, opt_pos, semi_pos, ws + 1, n);

  finalize_kernel<<<1, 1, 0, stream>>>(ws, out, 1.0f / (float)n);
}